// GCNLayer_14491219657349
// MI455X (gfx1250) — compile-verified
//
#include <hip/hip_runtime.h>

typedef float v2f __attribute__((ext_vector_type(2)));
typedef float v8f __attribute__((ext_vector_type(8)));

#define IN_DIM 64
#define OUT_DIM 64

// ---------------------------------------------------------------------------
// Kernel 1: zero the workspace (agg + deg). Grid-stride.
// ---------------------------------------------------------------------------
__global__ void gcn_zero_ws(float* __restrict__ p, long long n) {
  long long i = (long long)blockIdx.x * blockDim.x + threadIdx.x;
  long long stride = (long long)gridDim.x * blockDim.x;
  for (; i < n; i += stride) p[i] = 0.0f;
}

// ---------------------------------------------------------------------------
// Kernel 2: edge scatter.  thread -> (edge, col).  Consecutive threads cover
// consecutive cols of one feature row => coalesced 128B gathers per wave.
// agg (25.6MB) is L2-resident on MI455X (192MB L2), so the f32 atomics
// resolve at L2 atomic rate.  Lane col==0 also accumulates in-degree.
// ---------------------------------------------------------------------------
__global__ void gcn_edge_scatter(const float* __restrict__ feat,
                                 const float* __restrict__ w,
                                 const int* __restrict__ src,
                                 const int* __restrict__ dst,
                                 float* __restrict__ agg,
                                 float* __restrict__ deg,
                                 int n_edges) {
  long long idx = (long long)blockIdx.x * blockDim.x + threadIdx.x;
  long long total = (long long)n_edges * IN_DIM;
  if (idx >= total) return;
  int e   = (int)(idx >> 6);
  int col = (int)(idx & 63);
  int s = src[e];
  int d = dst[e];
  float m = feat[(long long)s * IN_DIM + col] * w[e];
  atomicAdd(&agg[(long long)d * IN_DIM + col], m);
  if (col == 0) atomicAdd(&deg[d], 1.0f);
}

// ---------------------------------------------------------------------------
// Kernel 3: degree-normalize + GEMM with V_WMMA_F32_16X16X4_F32 (full f32).
// Block = 128 threads = 4 wave32; block owns a 16-row M tile, wave wv owns
// N-cols [16*wv, 16*wv+16).  The normalized 16x64 h-tile is staged once in
// LDS and shared by all 4 waves (A operand).  K=64 -> 16 chained WMMAs.
//
// Fragment layouts (per CDNA5 ISA):
//  A 16x4 f32 (2 VGPR): lanes 0-15: row=lane, {a.x,a.y}={K0,K1};
//                       lanes 16-31: row=lane-16, {a.x,a.y}={K2,K3}
//  B  4x16 f32 (2 VGPR): lanes 0-15: col=lane,   {b.x,b.y}=rows{K0,K1};
//                        lanes 16-31: col=lane-16,{b.x,b.y}=rows{K2,K3}
//  C/D 16x16 f32 (8 VGPR): VGPR j: lanes 0-15 -> M=j, N=lane;
//                          lanes 16-31 -> M=j+8, N=lane-16
// Full tiles (the common case; N=100000 is a multiple of 16) take a
// branch-free store path; only a ragged final tile takes the guarded path.
// ---------------------------------------------------------------------------
__global__ void __launch_bounds__(128)
gcn_norm_gemm(const float* __restrict__ agg,
              const float* __restrict__ deg,
              const float* __restrict__ Wm,
              float* __restrict__ out,
              int n_nodes) {
  __shared__ float hs[16 * 64];

  const int  tid       = threadIdx.x;
  const int  row0      = blockIdx.x * 16;
  const bool full_tile = (row0 + 16) <= n_nodes;

  // Stage normalized h tile (16x64) into LDS: each thread writes 8 contiguous
  // floats of one row (8 threads per row).
  {
    int base = tid * 8;          // 0..1016
    int r    = base >> 6;        // 0..15
    int c    = base & 63;
    int grow = row0 + r;
    if (full_tile || grow < n_nodes) {
      float d    = deg[grow];
      float dinv = 1.0f / fmaxf(d, 1.0f);
      const float* rp = agg + (long long)grow * IN_DIM + c;
#pragma unroll
      for (int j = 0; j < 8; ++j) hs[base + j] = rp[j] * dinv;
    } else {
#pragma unroll
      for (int j = 0; j < 8; ++j) hs[base + j] = 0.0f;
    }
  }
  __syncthreads();

  const int lane = tid & 31;
  const int wv   = tid >> 5;       // 0..3 -> N tile
  const int hi   = lane >> 4;      // lane-half selects K pair
  const int l    = lane & 15;
  const int ncol = wv * 16 + l;

  v8f c = {};
#pragma unroll
  for (int k0 = 0; k0 < IN_DIM; k0 += 4) {
    const int ka = k0 + 2 * hi;
    v2f a, b;
    a.x = hs[l * IN_DIM + ka];
    a.y = hs[l * IN_DIM + ka + 1];
    b.x = Wm[ka * OUT_DIM + ncol];
    b.y = Wm[(ka + 1) * OUT_DIM + ncol];
    // 8 args: (neg_a, A, neg_b, B, c_mod, C, reuse_a, reuse_b)
    c = __builtin_amdgcn_wmma_f32_16x16x4_f32(
        false, a, false, b, (short)0, c, false, false);
  }

  // Store D: VGPR j -> row row0 + j + 8*hi, col ncol.
  float* outp = out + (long long)(row0 + 8 * hi) * OUT_DIM + ncol;
  if (full_tile) {
#pragma unroll
    for (int j = 0; j < 8; ++j) outp[j * OUT_DIM] = c[j];
  } else {
#pragma unroll
    for (int j = 0; j < 8; ++j) {
      if (row0 + j + 8 * hi < n_nodes) outp[j * OUT_DIM] = c[j];
    }
  }
}

// ---------------------------------------------------------------------------
// Launch: inputs in setup_inputs() order: features, w, W, src, dst.
// Workspace layout: agg[n_nodes*64] f32, then deg[n_nodes] f32 (~26MB).
// ---------------------------------------------------------------------------
extern "C" void kernel_launch(void* const* d_in, const int* in_sizes, int n_in,
                              void* d_out, int out_size, void* d_ws, size_t ws_size,
                              hipStream_t stream) {
  const float* feat = (const float*)d_in[0];
  const float* w    = (const float*)d_in[1];
  const float* Wm   = (const float*)d_in[2];
  const int*   src  = (const int*)d_in[3];
  const int*   dst  = (const int*)d_in[4];
  float*       outp = (float*)d_out;

  const int n_nodes = in_sizes[0] / IN_DIM;
  const int n_edges = in_sizes[1];

  float* agg = (float*)d_ws;
  float* deg = agg + (long long)n_nodes * IN_DIM;

  // 1) zero agg + deg
  long long zn = (long long)n_nodes * IN_DIM + n_nodes;
  gcn_zero_ws<<<2048, 256, 0, stream>>>(agg, zn);

  // 2) edge scatter (message + degree in one pass)
  long long total = (long long)n_edges * IN_DIM;
  int sblocks = (int)((total + 255) / 256);
  gcn_edge_scatter<<<sblocks, 256, 0, stream>>>(feat, w, src, dst, agg, deg, n_edges);

  // 3) normalize + GEMM (f32 WMMA)
  int mtiles = (n_nodes + 15) / 16;
  gcn_norm_gemm<<<mtiles, 128, 0, stream>>>(agg, deg, Wm, outp, n_nodes);
}